// _NSFFlowStack_66030827209327
// MI455X (gfx1250) — compile-verified
//
#include <hip/hip_runtime.h>
#include <cstdint>
#include <cstddef>

// ---------------- problem constants (fixed by the reference) ----------------
#define NLAYERS 4
#define KBINS   16
#define TBND    3.0f
#define MBW     0.001f
#define MBH     0.001f
#define MDER    0.001f
#define PPL     47                  // params per layer  (3*K - 1)
#define PPE     188                 // params per element (4 * 47)
#define N_B     32
#define N_T     512
#define N_D     32
#define N_TOTAL (N_B * N_T * N_D)   // 524288 elements
#define EPB     (N_T * N_D)         // 16384 elements per batch

// ---------------- tiling ----------------
#define ELEMS      64               // elements per block == threads per block (2 waves)
#define LDS_STRIDE 196              // padded dwords/element: mult of 4 (b128), %64==4 -> 2-way conflicts
#define NBLOCKS    (N_TOTAL / ELEMS)        // 8192
#define PART_PER_B (NBLOCKS / N_B)          // 256 partials per batch

// ---------------- CDNA5 async global->LDS staging ----------------
#if defined(__gfx1250__)
typedef int v4i __attribute__((vector_size(16)));
typedef __attribute__((address_space(1))) v4i* gptr_v4i;
typedef __attribute__((address_space(3))) v4i* lptr_v4i;

__device__ __forceinline__ void async_load_b128(const float* g, float* l) {
#if __has_builtin(__builtin_amdgcn_global_load_async_to_lds_b128)
  __builtin_amdgcn_global_load_async_to_lds_b128(
      (gptr_v4i)(uintptr_t)g,
      (lptr_v4i)l,
      /*offset=*/0, /*cpol=*/0);
#else
  unsigned lofs = (unsigned)(uintptr_t)(lptr_v4i)l;
  asm volatile("global_load_async_to_lds_b128 %0, %1, off"
               :: "v"(lofs), "v"(g) : "memory");
#endif
}
__device__ __forceinline__ void wait_async0() {
#if __has_builtin(__builtin_amdgcn_s_wait_asynccnt)
  __builtin_amdgcn_s_wait_asynccnt(0);
#else
  asm volatile("s_wait_asynccnt 0" ::: "memory");
#endif
}
#else
// host pass / other arch: plain copy so the source still parses & links
__device__ __forceinline__ void async_load_b128(const float* g, float* l) {
  l[0] = g[0]; l[1] = g[1]; l[2] = g[2]; l[3] = g[3];
}
__device__ __forceinline__ void wait_async0() {}
#endif

__device__ __forceinline__ float softplusf(float v) {
  // matches jax.nn.softplus: max(v,0) + log1p(exp(-|v|))
  return fmaxf(v, 0.0f) + log1pf(expf(-fabsf(v)));
}

// ---------------- main kernel: 1 element per thread, params staged via async LDS ----
extern "C" __global__ __launch_bounds__(ELEMS)
void nsf_main(const float* __restrict__ params, const float* __restrict__ x,
              float* __restrict__ y, float* __restrict__ partial) {
  __shared__ float lds[ELEMS * LDS_STRIDE];   // 64*196*4 = 50,176 B

  const int tid   = threadIdx.x;
  const int blk   = blockIdx.x;
  const size_t e0 = (size_t)blk * ELEMS;

  // ---- stage 64 x 188 f32 tile: 47 x b128 chunks per element, coalesced ----
  {
    const float* gbase = params + e0 * PPE;
    #pragma unroll
    for (int j = tid; j < ELEMS * PPL; j += ELEMS) {   // 47 iterations
      const int e = j / PPL;
      const int c = j - e * PPL;
      async_load_b128(gbase + (size_t)e * PPE + (size_t)c * 4,
                      &lds[e * LDS_STRIDE + c * 4]);
    }
  }
  wait_async0();
  __syncthreads();

  float xv = x[e0 + tid];
  float ld_acc = 0.0f;

  const float cst    = logf(expm1f(1.0f - MDER));       // pad value for derivatives
  const float d_edge = MDER + softplusf(cst);           // == 1.0 up to rounding
  const float CWF    = 1.0f - MBW * (float)KBINS;
  const float CHF    = 1.0f - MBH * (float)KBINS;

  #pragma unroll 1
  for (int li = 0; li < NLAYERS; ++li) {
    const float* p = &lds[tid * LDS_STRIDE + li * PPL];

    float uw[KBINS], uh[KBINS], ud[KBINS - 1];
    #pragma unroll
    for (int i = 0; i < KBINS; ++i)     uw[i] = p[i];
    #pragma unroll
    for (int i = 0; i < KBINS; ++i)     uh[i] = p[KBINS + i];
    #pragma unroll
    for (int i = 0; i < KBINS - 1; ++i) ud[i] = p[2 * KBINS + i];

    // stable softmax for widths
    float mw = uw[0];
    #pragma unroll
    for (int i = 1; i < KBINS; ++i) mw = fmaxf(mw, uw[i]);
    float ew[KBINS]; float sw = 0.0f;
    #pragma unroll
    for (int i = 0; i < KBINS; ++i) { ew[i] = expf(uw[i] - mw); sw += ew[i]; }
    const float isw = 1.0f / sw;

    // stable softmax for heights
    float mh = uh[0];
    #pragma unroll
    for (int i = 1; i < KBINS; ++i) mh = fmaxf(mh, uh[i]);
    float eh[KBINS]; float sh = 0.0f;
    #pragma unroll
    for (int i = 0; i < KBINS; ++i) { eh[i] = expf(uh[i] - mh); sh += eh[i]; }
    const float ish = 1.0f / sh;

    // knot positions (all constant-indexed -> stays in VGPRs)
    float cw[KBINS + 1], ch[KBINS + 1], dv[KBINS + 1];
    cw[0] = -TBND; ch[0] = -TBND;
    float aw = 0.0f, ah = 0.0f;
    #pragma unroll
    for (int i = 0; i < KBINS; ++i) {
      aw += MBW + CWF * ew[i] * isw;
      ah += MBH + CHF * eh[i] * ish;
      cw[i + 1] = 2.0f * TBND * aw - TBND;
      ch[i + 1] = 2.0f * TBND * ah - TBND;
    }
    cw[KBINS] = TBND; ch[KBINS] = TBND;
    dv[0] = d_edge; dv[KBINS] = d_edge;
    #pragma unroll
    for (int i = 1; i < KBINS; ++i) dv[i] = MDER + softplusf(ud[i - 1]);

    const float xc = fminf(fmaxf(xv, -TBND), TBND);

    // bin select: last j with xc >= cw[j]  (== clip(sum(xc>=cw)-1, 0, K-1))
    float xk = cw[0], wk = cw[1] - cw[0];
    float yk = ch[0], hk = ch[1] - ch[0];
    float dk = dv[0], dk1 = dv[1];
    #pragma unroll
    for (int j = 1; j < KBINS; ++j) {
      const bool t = (xc >= cw[j]);
      xk  = t ? cw[j]             : xk;
      wk  = t ? (cw[j+1] - cw[j]) : wk;
      yk  = t ? ch[j]             : yk;
      hk  = t ? (ch[j+1] - ch[j]) : hk;
      dk  = t ? dv[j]             : dk;
      dk1 = t ? dv[j+1]           : dk1;
    }

    const float delta = hk / wk;
    const float theta = (xc - xk) / wk;
    const float omt   = 1.0f - theta;
    const float t1m   = theta * omt;
    const float th2   = theta * theta;
    const float denom = delta + (dk + dk1 - 2.0f * delta) * t1m;
    const float yv    = yk + hk * (delta * th2 + dk * t1m) / denom;
    const float dnum  = delta * delta * (dk1 * th2 + 2.0f * delta * t1m + dk * omt * omt);
    const float ldv   = logf(dnum) - 2.0f * logf(denom);

    const bool inside = (xv >= -TBND) && (xv <= TBND);
    xv     = inside ? yv : xv;
    ld_acc += inside ? ldv : 0.0f;
  }

  y[e0 + tid] = xv;

  // ---- deterministic block reduction of logdet contribution (2 waves) ----
  #pragma unroll
  for (int off = 16; off > 0; off >>= 1)
    ld_acc += __shfl_down(ld_acc, off, 32);
  __shared__ float wsum[ELEMS / 32];
  if ((tid & 31) == 0) wsum[tid >> 5] = ld_acc;
  __syncthreads();
  if (tid == 0) {
    float s = 0.0f;
    #pragma unroll
    for (int w = 0; w < ELEMS / 32; ++w) s += wsum[w];
    partial[blk] = s;
  }
}

// ---------------- second pass: fixed-order reduction of per-block partials ----------
extern "C" __global__ __launch_bounds__(256)
void nsf_reduce(const float* __restrict__ partial, float* __restrict__ logdet) {
  const int b   = blockIdx.x;     // one block per batch
  const int tid = threadIdx.x;
  float v = 0.0f;
  for (int i = tid; i < PART_PER_B; i += 256)
    v += partial[(size_t)b * PART_PER_B + i];
  #pragma unroll
  for (int off = 16; off > 0; off >>= 1)
    v += __shfl_down(v, off, 32);
  __shared__ float wsum[8];
  if ((tid & 31) == 0) wsum[tid >> 5] = v;
  __syncthreads();
  if (tid == 0) {
    float s = 0.0f;
    #pragma unroll
    for (int w = 0; w < 8; ++w) s += wsum[w];
    logdet[b] = s;
  }
}

// ---------------- entry point ----------------
extern "C" void kernel_launch(void* const* d_in, const int* in_sizes, int n_in,
                              void* d_out, int out_size, void* d_ws, size_t ws_size,
                              hipStream_t stream) {
  (void)in_sizes; (void)n_in; (void)out_size; (void)ws_size;
  const float* params = (const float*)d_in[0];   // [32,512,32,188] f32
  const float* x      = (const float*)d_in[1];   // [32,512,32]     f32
  float* y      = (float*)d_out;                 // first 524288 outputs
  float* logdet = (float*)d_out + N_TOTAL;       // last 32 outputs
  float* part   = (float*)d_ws;                  // 8192 f32 partials

  hipLaunchKernelGGL(nsf_main, dim3(NBLOCKS), dim3(ELEMS), 0, stream,
                     params, x, y, part);
  hipLaunchKernelGGL(nsf_reduce, dim3(N_B), dim3(256), 0, stream,
                     part, logdet);
}